// NetSoNTopSIN_20366734917783
// MI455X (gfx1250) — compile-verified
//
#include <hip/hip_runtime.h>
#include <hip/hip_bf16.h>
#include <float.h>

typedef __attribute__((ext_vector_type(2))) float v2f;
typedef __attribute__((ext_vector_type(4))) float f32x4;
typedef __attribute__((ext_vector_type(8))) float v8f;

#define HW    50176      // 224*224
#define BATCH 128
#define CH    33
#define GRP   100
#define OUTC  10
#define KPAD  36         // 33 padded to multiple of 4
#define MT    8          // 128/16 m-tiles
#define NT    7          // 112/16 n-tiles (100 padded)

// ---------------------------------------------------------------------------
// Kernel 1: global average pool over H*W for each (b,c). Bandwidth-bound:
// 847 MB streamed once -> non-temporal 128-bit loads, fully coalesced.
// One block per (b,c) row: 50176 floats = 256 threads * 49 float4 exactly.
// Writes x_sun into d_out and tanh(x_sun) into workspace for the WMMA GEMM.
// ---------------------------------------------------------------------------
__global__ __launch_bounds__(256) void k_pool(const float* __restrict__ maps,
                                              float* __restrict__ out_xsun,
                                              float* __restrict__ tanhA) {
  const int row = blockIdx.x;                 // row = b*33 + c
  const int t   = threadIdx.x;
  const f32x4* p = (const f32x4*)maps + (size_t)row * (HW / 4);

  f32x4 acc = {0.f, 0.f, 0.f, 0.f};
#pragma unroll 7
  for (int i = 0; i < 49; ++i) {
    f32x4 v = __builtin_nontemporal_load(&p[i * 256 + t]);   // streaming NT load
    acc.x += v.x; acc.y += v.y; acc.z += v.z; acc.w += v.w;
  }
  float s = (acc.x + acc.y) + (acc.z + acc.w);

  // wave32 shuffle reduce
#pragma unroll
  for (int off = 16; off > 0; off >>= 1) s += __shfl_down(s, off, 32);

  __shared__ float wsum[8];
  if ((t & 31) == 0) wsum[t >> 5] = s;
  __syncthreads();
  if (t == 0) {
    float tot = 0.f;
#pragma unroll
    for (int w = 0; w < 8; ++w) tot += wsum[w];
    float mean = tot * (1.0f / (float)HW);
    out_xsun[row] = mean;          // x_sun output slice
    tanhA[row]    = tanhf(mean);   // fc1 input
  }
}

// ---------------------------------------------------------------------------
// Kernel 2: x_groups = relu(tanh(x_sun) @ w1.T) via V_WMMA_F32_16X16X4_F32.
// One wave per 16x16 output tile; 9 chained K=4 WMMA steps (K padded to 36).
// Fragment loads are BRANCHLESS: indices are clamped so every load is legal,
// and out-of-range lanes are zeroed with selects (v_cndmask, no exec forks).
// A layout (ISA 7.12.2, 32-bit A 16x4): lane L -> M = L&15,
//   VGPR0 = K(+0 or +2 by lane half), VGPR1 = K(+1 or +3).
// B (4x16) mirrored: lane L -> N = L&15, same K-half split.
// C/D: lane L -> N = L&15; VGPR r -> M = r (+8 for lanes 16-31).
// EXEC is all-1s at every WMMA.
// ---------------------------------------------------------------------------
__global__ __launch_bounds__(32) void k_gemm_wmma(const float* __restrict__ tanhA,
                                                  const float* __restrict__ w1,
                                                  float* __restrict__ xg) {
  const int tile  = blockIdx.x;
  const int m0    = (tile / NT) * 16;
  const int n0    = (tile % NT) * 16;
  const int lane  = threadIdx.x;
  const int l16   = lane & 15;
  const int khalf = (lane >> 4) << 1;     // 0 for lanes 0-15, 2 for lanes 16-31
  const int m     = m0 + l16;             // always < 128
  const int n     = n0 + l16;             // may be >= 100 (padded)

  const bool ncol   = (n < GRP);
  const int  nclamp = ncol ? n : (GRP - 1);          // safe row of w1
  const float* __restrict__ arow = tanhA + m * CH;
  const float* __restrict__ brow = w1 + nclamp * CH;

  // Preload all 9 K-steps branchlessly (2 elems each for A and B per lane).
  v2f afrag[KPAD / 4];
  v2f bfrag[KPAD / 4];
#pragma unroll
  for (int ks = 0; ks < KPAD / 4; ++ks) {
    const int ka  = ks * 4 + khalf;
    const int k0c = (ka     < CH) ? ka     : (CH - 1);  // clamp -> legal load
    const int k1c = (ka + 1 < CH) ? ka + 1 : (CH - 1);
    const float a0 = arow[k0c];
    const float a1 = arow[k1c];
    const float b0 = brow[k0c];
    const float b1 = brow[k1c];
    afrag[ks].x = (ka     < CH) ? a0 : 0.f;
    afrag[ks].y = (ka + 1 < CH) ? a1 : 0.f;
    bfrag[ks].x = (ncol && ka     < CH) ? b0 : 0.f;     // B[k][n] = w1[n][k]
    bfrag[ks].y = (ncol && ka + 1 < CH) ? b1 : 0.f;
  }

  v8f acc = {};
#pragma unroll
  for (int ks = 0; ks < KPAD / 4; ++ks) {
    acc = __builtin_amdgcn_wmma_f32_16x16x4_f32(
        /*neg_a=*/false, afrag[ks], /*neg_b=*/false, bfrag[ks],
        /*c_mod=*/(short)0, acc, /*reuse_a=*/false, /*reuse_b=*/false);
  }

  const int mbase = m0 + ((lane >> 4) << 3);   // +0 or +8
  if (ncol) {
#pragma unroll
    for (int r = 0; r < 8; ++r) {
      float v = acc[r];
      xg[(mbase + r) * GRP + n] = v > 0.f ? v : 0.f;   // fused ReLU
    }
  }
}

// ---------------------------------------------------------------------------
// Kernel 3: vote = x_groups[b,i]*w2[o,i]; top-k cumulative sums for
// k in {3,4,5,6,7,10,15,20} plus the dense matmul row. One thread per (b,o):
// sorted top-20 insertion list over 100 values, then cumsum reads.
// ---------------------------------------------------------------------------
__global__ __launch_bounds__(256) void k_topk(const float* __restrict__ xg,
                                              const float* __restrict__ w2,
                                              float* __restrict__ out) {
  const int t = blockIdx.x * 256 + threadIdx.x;
  if (t >= BATCH * OUTC) return;
  const int b = t / OUTC, o = t % OUTC;
  const float* __restrict__ x = xg + b * GRP;
  const float* __restrict__ w = w2 + o * GRP;

  float top[20];
#pragma unroll
  for (int j = 0; j < 20; ++j) top[j] = -FLT_MAX;

  float dense = 0.f;
  for (int i = 0; i < GRP; ++i) {
    const float v = x[i] * w[i];
    dense += v;
    if (v > top[19]) {                    // insert keeping descending order
      int j = 19;
      while (j > 0 && top[j - 1] < v) { top[j] = top[j - 1]; --j; }
      top[j] = v;
    }
  }

  float cs[20];
  float s = 0.f;
#pragma unroll
  for (int j = 0; j < 20; ++j) { s += top[j]; cs[j] = s; }

  float* son = out + BATCH * CH;          // after x_sun (4224 floats)
  const int stride = BATCH * OUTC;        // 1280 per k-slice
  son[0 * stride + t] = cs[2];            // k=3
  son[1 * stride + t] = cs[3];            // k=4
  son[2 * stride + t] = cs[4];            // k=5
  son[3 * stride + t] = cs[5];            // k=6
  son[4 * stride + t] = cs[6];            // k=7
  son[5 * stride + t] = cs[9];            // k=10
  son[6 * stride + t] = cs[14];           // k=15
  son[7 * stride + t] = cs[19];           // k=20
  son[8 * stride + t] = dense;            // dense pass
}

extern "C" void kernel_launch(void* const* d_in, const int* in_sizes, int n_in,
                              void* d_out, int out_size, void* d_ws, size_t ws_size,
                              hipStream_t stream) {
  const float* maps = (const float*)d_in[0];   // [128,33,224,224]
  const float* w1   = (const float*)d_in[1];   // [100,33]
  const float* w2   = (const float*)d_in[2];   // [10,100]
  float* out = (float*)d_out;                  // 4224 + 9*1280 floats

  float* tanhA = (float*)d_ws;                 // [128,33]
  float* xg    = tanhA + BATCH * CH;           // [128,100]

  k_pool<<<BATCH * CH, 256, 0, stream>>>(maps, out, tanhA);
  k_gemm_wmma<<<MT * NT, 32, 0, stream>>>(tanhA, w1, xg);
  k_topk<<<(BATCH * OUTC + 255) / 256, 256, 0, stream>>>(xg, w2, out);
}